// Retina_4887672783521
// MI455X (gfx1250) — compile-verified
//
#include <hip/hip_runtime.h>
#include <stdint.h>

// Problem constants (match reference)
#define NB   64    // batch
#define CCH  3     // channels
#define HW_  512   // H == W
#define GSZ  64    // glimpse size
#define NK   3     // number of scales

// LDS staging strip: up to 32 input rows (8 output rows * m<=4) x 256 cols
#define LROWS 32
#define LCOLS 256

// ---- gfx1250 async data mover: global -> LDS, tracked by ASYNCcnt ----
__device__ __forceinline__ void async_ld_f32(uint32_t lds_byte_off, const float* g) {
    uint64_t ga = (uint64_t)(uintptr_t)g;
    // GV mode: vdst = per-lane LDS byte address, vaddr = 64-bit global address
    asm volatile("global_load_async_to_lds_b32 %0, %1, off"
                 :: "v"(lds_byte_off), "v"(ga)
                 : "memory");
}

__device__ __forceinline__ void wait_async0() {
    asm volatile("s_wait_asynccnt 0" ::: "memory");
}

__global__ __launch_bounds__(256)
void retina_glimpse_kernel(const float* __restrict__ x,
                           const float* __restrict__ l,
                           float* __restrict__ out) {
    __shared__ float smem[LROWS * LCOLS];   // 32 KB

    const int bx  = blockIdx.x;             // 0 .. B*K*C-1
    const int b   = bx / (NK * CCH);
    const int rem = bx % (NK * CCH);
    const int k   = rem / CCH;              // scale index
    const int c   = rem % CCH;              // channel

    const int m    = 1 << k;                // 1, 2, 4
    const int size = GSZ * m;               // 64, 128, 256
    const int pad  = size >> 1;

    // denormalize: start = int(0.5 * ((l + 1) * H)) ; trunc == floor (value >= 0)
    const float lx = l[2 * b + 0];
    const float ly = l[2 * b + 1];
    const int sx = (int)(0.5f * ((lx + 1.0f) * (float)HW_));
    const int sy = (int)(0.5f * ((ly + 1.0f) * (float)HW_));

    const int col0 = sx - pad;              // image col of patch col 0
    const int row0 = sy - pad;              // image row of patch row 0

    const int tid = (int)threadIdx.x;       // 0..255
    const int ox  = tid & 63;               // output column
    const int sub = tid >> 6;               // 0..3 (output row within group of 4)

    const float* xbc = x + (size_t)(b * CCH + c) * (size_t)(HW_ * HW_);
    float* obase = out + ((size_t)b * (NK * CCH) + (size_t)(k * CCH + c)) * (GSZ * GSZ);

    const float inv = 1.0f / (float)(m * m);
    const int nr = 8 * m;                   // input rows staged per chunk (8 output rows)

    const uint32_t lds_base = (uint32_t)(uintptr_t)(&smem[0]);

    for (int chunk = 0; chunk < 8; ++chunk) {
        // 1) zero-fill the strip (implements the zero padding / clipping)
        const int nfill = nr * LCOLS;
        for (int i = tid; i < nfill; i += 256) smem[i] = 0.0f;
        __syncthreads();

        // 2) async-stage the valid window of the needed input rows into LDS.
        //    lane j = tid covers patch column j (j < size).
        const int rbase = row0 + chunk * nr;
        if (tid < size) {
            const int cc = col0 + tid;
            if (cc >= 0 && cc < HW_) {
                for (int lr = 0; lr < nr; ++lr) {
                    const int r = rbase + lr;
                    if (r >= 0 && r < HW_) {
                        async_ld_f32(lds_base + (uint32_t)((lr * LCOLS + tid) * 4),
                                     xbc + (size_t)r * HW_ + cc);
                    }
                }
            }
        }
        wait_async0();          // this wave's async LDS writes are complete
        __syncthreads();        // all waves' writes visible to everyone

        // 3) block-mean pool m x m from LDS -> 8 output rows
        #pragma unroll
        for (int it = 0; it < 2; ++it) {
            const int gl = it * 4 + sub;                 // local output row 0..7
            float s = 0.0f;
            for (int a = 0; a < m; ++a) {
                const float* lrow = &smem[(gl * m + a) * LCOLS + ox * m];
                for (int bb = 0; bb < m; ++bb) s += lrow[bb];
            }
            const int gy = chunk * 8 + gl;
            obase[gy * GSZ + ox] = s * inv;
        }
        __syncthreads();        // strip reusable next chunk
    }
}

extern "C" void kernel_launch(void* const* d_in, const int* in_sizes, int n_in,
                              void* d_out, int out_size, void* d_ws, size_t ws_size,
                              hipStream_t stream) {
    (void)in_sizes; (void)n_in; (void)d_ws; (void)ws_size; (void)out_size;
    const float* x = (const float*)d_in[0];   // [64,3,512,512] f32
    const float* l = (const float*)d_in[1];   // [64,2] f32
    float* out = (float*)d_out;               // [64, 9*64*64] f32

    const int nblocks = NB * NK * CCH;        // 576
    retina_glimpse_kernel<<<nblocks, 256, 0, stream>>>(x, l, out);
}